// SelectiveSSM_19756849561708
// MI455X (gfx1250) — compile-verified
//
#include <hip/hip_runtime.h>
#include <hip/hip_bf16.h>
#include <math.h>

// Problem dims (fixed by the reference)
#define BB 2
#define LL 2048
#define DD 1024
#define NN 16
#define RR 64          // DT_RANK
#define CH 16          // scan chunks
#define CLEN (LL / CH) // 128

typedef float v2f __attribute__((ext_vector_type(2)));
typedef float v8f __attribute__((ext_vector_type(8)));
typedef int   v4i __attribute__((vector_size(16)));   // builtin's pointee type

__device__ __forceinline__ float silu_f(float v) {
    return v / (1.0f + __expf(-v));
}
__device__ __forceinline__ float softplus_f(float v) {
    return (v > 20.0f) ? v : log1pf(__expf(v));
}

// ---------------------------------------------------------------------------
// Async global->LDS staging (CDNA5 GLOBAL_LOAD_ASYNC_TO_LDS_B128, ASYNCcnt),
// with a synchronous fallback so compilation never depends on the builtin.
// ---------------------------------------------------------------------------
#if defined(__HIP_DEVICE_COMPILE__)
#if __has_builtin(__builtin_amdgcn_global_load_async_to_lds_b128) && \
    __has_builtin(__builtin_amdgcn_s_wait_asynccnt)
#define USE_ASYNC_COPY 1
#endif
#endif

__device__ __forceinline__ void copy16_to_lds(float* dst_lds, const float* src_glb) {
#ifdef USE_ASYNC_COPY
    __builtin_amdgcn_global_load_async_to_lds_b128(
        (v4i*)src_glb, (v4i*)dst_lds, 0, 0);
#else
    *(float4*)dst_lds = *(const float4*)src_glb;
#endif
}
__device__ __forceinline__ void copy_wait() {
#ifdef USE_ASYNC_COPY
    __builtin_amdgcn_s_wait_asynccnt(0);
#endif
}

// ---------------------------------------------------------------------------
// Kernel 1: depthwise conv(3, SAME, no bias) + SiLU  ->  u [B, L, D]
// ---------------------------------------------------------------------------
__global__ void conv_silu_kernel(const float* __restrict__ x,
                                 const float* __restrict__ kern,
                                 float* __restrict__ u) {
    int idx = blockIdx.x * 256 + threadIdx.x;      // 0 .. B*L*D/4-1
    int d4  = (idx & 255) * 4;
    int l   = (idx >> 8) & (LL - 1);
    int b   = idx >> 19;

    size_t base = ((size_t)b * LL + l) * DD + d4;
    float4 xc = *(const float4*)(x + base);
    float4 xm = make_float4(0.f, 0.f, 0.f, 0.f);
    float4 xp = make_float4(0.f, 0.f, 0.f, 0.f);
    if (l > 0)      xm = *(const float4*)(x + base - DD);
    if (l < LL - 1) xp = *(const float4*)(x + base + DD);

    float4 k0 = *(const float4*)(kern + 0 * DD + d4);
    float4 k1 = *(const float4*)(kern + 1 * DD + d4);
    float4 k2 = *(const float4*)(kern + 2 * DD + d4);

    float4 r;
    r.x = silu_f(fmaf(k0.x, xm.x, fmaf(k1.x, xc.x, k2.x * xp.x)));
    r.y = silu_f(fmaf(k0.y, xm.y, fmaf(k1.y, xc.y, k2.y * xp.y)));
    r.z = silu_f(fmaf(k0.z, xm.z, fmaf(k1.z, xc.z, k2.z * xp.z)));
    r.w = silu_f(fmaf(k0.w, xm.w, fmaf(k1.w, xc.w, k2.w * xp.w)));
    *(float4*)(u + base) = r;
}

// ---------------------------------------------------------------------------
// Kernel 2: fused projections via V_WMMA_F32_16X16X4_F32.
//   Bc,Cc = u @ W_bc + b_bc   (wave 0,1)   dtlow = u @ W_dt + b_dt (wave 2..5)
// LDS stride 132: 16B-aligned rows for B128 async stores, conflict-free
// column reads (bank = 4*row + k).
// ---------------------------------------------------------------------------
#define KC 128
#define ASTRIDE 132

__global__ void proj_bc_dt_kernel(const float* __restrict__ u,
                                  const float* __restrict__ W_bc,
                                  const float* __restrict__ b_bc,
                                  const float* __restrict__ W_dt,
                                  const float* __restrict__ b_dt,
                                  float* __restrict__ dBc,
                                  float* __restrict__ dCc,
                                  float* __restrict__ dtlow) {
    __shared__ float sA[16 * ASTRIDE];

    const int tid   = threadIdx.x;
    const int wave  = tid >> 5;          // 0..5
    const int lane  = tid & 31;
    const int col   = lane & 15;
    const int hi    = lane >> 4;         // 0: K=0/1 half, 1: K=2/3 half
    const int mbase = blockIdx.x * 16;   // 256 blocks cover 4096 rows

    const float* W;  const float* bias;  int ldw, nb;
    if (wave < 2) { W = W_bc; bias = b_bc; ldw = 2 * NN; nb = wave * 16; }
    else          { W = W_dt; bias = b_dt; ldw = RR;     nb = (wave - 2) * 16; }

    v8f acc = {};
    for (int kc = 0; kc < DD; kc += KC) {
        __syncthreads();
        // 16 rows x 128 cols = 512 float4s, 192 threads
        for (int i = tid; i < 512; i += 192) {
            int r = i >> 5, c4 = (i & 31) * 4;
            copy16_to_lds(&sA[r * ASTRIDE + c4],
                          &u[(size_t)(mbase + r) * DD + kc + c4]);
        }
        copy_wait();
        __syncthreads();

        #pragma unroll 8
        for (int k = 0; k < KC; k += 4) {
            const int kk = k + (hi ? 2 : 0);
            v2f a, bf;
            a.x = sA[col * ASTRIDE + kk];
            a.y = sA[col * ASTRIDE + kk + 1];
            const int kg = kc + kk;
            bf.x = W[(size_t)kg * ldw + nb + col];
            bf.y = W[(size_t)(kg + 1) * ldw + nb + col];
            acc = __builtin_amdgcn_wmma_f32_16x16x4_f32(
                false, a, false, bf, (short)0, acc, false, false);
        }
    }

    const float bv = bias[nb + col];
    #pragma unroll
    for (int r = 0; r < 8; ++r) {
        const int row = mbase + r + (hi ? 8 : 0);
        const float v = acc[r] + bv;
        if (wave == 0)      dBc[(size_t)row * NN + col] = v;
        else if (wave == 1) dCc[(size_t)row * NN + col] = v;
        else                dtlow[(size_t)row * RR + (wave - 2) * 16 + col] = v;
    }
}

// ---------------------------------------------------------------------------
// Kernel 3: dt = softplus(dtlow @ W_dtp + b_dtp), [4096 x 64] @ [64 x 1024].
// ---------------------------------------------------------------------------
__global__ void dt_proj_kernel(const float* __restrict__ dtlow,
                               const float* __restrict__ W_dtp,
                               const float* __restrict__ b_dtp,
                               float* __restrict__ dt) {
    __shared__ float sA[16 * 68];

    const int tid   = threadIdx.x;
    const int mtile = blockIdx.x >> 4;
    const int cg    = blockIdx.x & 15;
    const int mbase = mtile * 16;

    // 16 rows x 64 cols = 256 float4s, 128 threads
    for (int i = tid; i < 256; i += 128) {
        int r = i >> 4, c4 = (i & 15) * 4;
        copy16_to_lds(&sA[r * 68 + c4],
                      &dtlow[(size_t)(mbase + r) * RR + c4]);
    }
    copy_wait();
    __syncthreads();

    const int wave = tid >> 5;
    const int lane = tid & 31;
    const int col  = lane & 15;
    const int hi   = lane >> 4;
    const int nb   = (cg * 4 + wave) * 16;

    v8f acc = {};
    #pragma unroll
    for (int k = 0; k < RR; k += 4) {
        const int kk = k + (hi ? 2 : 0);
        v2f a, bf;
        a.x = sA[col * 68 + kk];
        a.y = sA[col * 68 + kk + 1];
        bf.x = W_dtp[(size_t)kk * DD + nb + col];
        bf.y = W_dtp[(size_t)(kk + 1) * DD + nb + col];
        acc = __builtin_amdgcn_wmma_f32_16x16x4_f32(
            false, a, false, bf, (short)0, acc, false, false);
    }

    const float bv = b_dtp[nb + col];
    #pragma unroll
    for (int r = 0; r < 8; ++r) {
        const int row = mbase + r + (hi ? 8 : 0);
        dt[(size_t)row * DD + nb + col] = softplus_f(acc[r] + bv);
    }
}

// ---------------------------------------------------------------------------
// Chunked selective scan (linear recurrence h' = alpha*h + beta):
//  pass1: per (b,d,n,chunk) compute G = prod(alpha), H = local end state
//  pass2: per (b,d,n) 16-step prefix over chunks -> h0 per chunk
//  pass3: rescan within chunk from h0, shuffle-reduce Sum_n C*h, + D skip
// Index layout for G/H/h0: gid = ((b*D + d)*CH + c)*N + n.
// ---------------------------------------------------------------------------
__global__ void scan_pass1_kernel(const float* __restrict__ x,
                                  const float* __restrict__ dt,
                                  const float* __restrict__ Bc,
                                  const float* __restrict__ A_log,
                                  float* __restrict__ G,
                                  float* __restrict__ H) {
    const int gid = blockIdx.x * 256 + threadIdx.x;   // B*D*CH*N
    const int n   = gid & (NN - 1);
    const int c   = (gid >> 4) & (CH - 1);
    const int d   = (gid >> 8) & (DD - 1);
    const int b   = gid >> 18;

    const float Ac = -__expf(A_log[d * NN + n]);
    const int   l0 = c * CLEN;
    const float* dtp = dt + ((size_t)b * LL + l0) * DD + d;
    const float* xp  = x  + ((size_t)b * LL + l0) * DD + d;
    const float* Bp  = Bc + ((size_t)b * LL + l0) * NN + n;

    float g = 1.0f, h = 0.0f;
    for (int l = 0; l < CLEN; ++l) {
        const float dtv = dtp[(size_t)l * DD];
        const float xv  = xp[(size_t)l * DD];
        const float bv  = Bp[(size_t)l * NN];
        const float alpha = __expf(Ac * dtv);
        h = fmaf(alpha, h, bv * xv * dtv);
        g *= alpha;
    }
    G[gid] = g;
    H[gid] = h;
}

__global__ void scan_pass2_kernel(const float* __restrict__ G,
                                  const float* __restrict__ H,
                                  float* __restrict__ h0) {
    const int gid = blockIdx.x * 256 + threadIdx.x;   // B*D*N
    const int n   = gid & (NN - 1);
    const int bd  = gid >> 4;
    const size_t base = (size_t)bd * CH * NN + n;

    float h = 0.0f;
    #pragma unroll
    for (int c = 0; c < CH; ++c) {
        const size_t o = base + (size_t)c * NN;
        h0[o] = h;
        h = fmaf(G[o], h, H[o]);   // state after chunk c = G_c*h0_c + H_c
    }
}

__global__ void scan_pass3_kernel(const float* __restrict__ x,
                                  const float* __restrict__ dt,
                                  const float* __restrict__ Bc,
                                  const float* __restrict__ Cc,
                                  const float* __restrict__ A_log,
                                  const float* __restrict__ Dcoeff,
                                  const float* __restrict__ h0,
                                  float* __restrict__ y) {
    const int gid = blockIdx.x * 256 + threadIdx.x;   // B*D*CH*N
    const int n   = gid & (NN - 1);
    const int c   = (gid >> 4) & (CH - 1);
    const int d   = (gid >> 8) & (DD - 1);
    const int b   = gid >> 18;

    const float Ac = -__expf(A_log[d * NN + n]);
    const float Dc = Dcoeff[d];
    const int   l0 = c * CLEN;
    const float* dtp = dt + ((size_t)b * LL + l0) * DD + d;
    const float* xp  = x  + ((size_t)b * LL + l0) * DD + d;
    const float* Bp  = Bc + ((size_t)b * LL + l0) * NN + n;
    const float* Cp  = Cc + ((size_t)b * LL + l0) * NN + n;
    float*       yp  = y  + ((size_t)b * LL + l0) * DD + d;

    float h = h0[gid];
    for (int l = 0; l < CLEN; ++l) {
        const float dtv = dtp[(size_t)l * DD];
        const float xv  = xp[(size_t)l * DD];
        const float bv  = Bp[(size_t)l * NN];
        const float cv  = Cp[(size_t)l * NN];

        const float alpha = __expf(Ac * dtv);
        h = fmaf(alpha, h, bv * xv * dtv);

        float contrib = cv * h;
        contrib += __shfl_xor(contrib, 8, 32);
        contrib += __shfl_xor(contrib, 4, 32);
        contrib += __shfl_xor(contrib, 2, 32);
        contrib += __shfl_xor(contrib, 1, 32);
        if (n == 0) yp[(size_t)l * DD] = fmaf(xv, Dc, contrib);
    }
}

// ---------------------------------------------------------------------------
extern "C" void kernel_launch(void* const* d_in, const int* in_sizes, int n_in,
                              void* d_out, int out_size, void* d_ws, size_t ws_size,
                              hipStream_t stream) {
    const float* x      = (const float*)d_in[0];
    const float* conv_k = (const float*)d_in[1];
    const float* A_log  = (const float*)d_in[2];
    const float* W_bc   = (const float*)d_in[3];
    const float* b_bc   = (const float*)d_in[4];
    const float* Dco    = (const float*)d_in[5];
    const float* W_dt   = (const float*)d_in[6];
    const float* b_dt   = (const float*)d_in[7];
    const float* W_dtp  = (const float*)d_in[8];
    const float* b_dtp  = (const float*)d_in[9];
    float* y = (float*)d_out;

    // Workspace (floats): u | dt | Bc | Cc | dtlow | G | H | h0  (~40 MB)
    float* ws    = (float*)d_ws;
    float* u     = ws;
    float* dt    = u     + (size_t)BB * LL * DD;
    float* dBc   = dt    + (size_t)BB * LL * DD;
    float* dCc   = dBc   + (size_t)BB * LL * NN;
    float* dtlow = dCc   + (size_t)BB * LL * NN;
    float* G     = dtlow + (size_t)BB * LL * RR;
    float* H     = G     + (size_t)BB * DD * NN * CH;
    float* h0    = H     + (size_t)BB * DD * NN * CH;

    // 1) conv + silu
    conv_silu_kernel<<<4096, 256, 0, stream>>>(x, conv_k, u);

    // 2) Bc/Cc/dtlow projections (WMMA f32)
    proj_bc_dt_kernel<<<256, 192, 0, stream>>>(u, W_bc, b_bc, W_dt, b_dt,
                                               dBc, dCc, dtlow);

    // 3) dt projection + softplus (WMMA f32)
    dt_proj_kernel<<<4096, 128, 0, stream>>>(dtlow, W_dtp, b_dtp, dt);

    // 4) chunked scan: 16x more scan parallelism than single pass
    scan_pass1_kernel<<<2048, 256, 0, stream>>>(x, dt, dBc, A_log, G, H);
    scan_pass2_kernel<<<128, 256, 0, stream>>>(G, H, h0);
    scan_pass3_kernel<<<2048, 256, 0, stream>>>(x, dt, dBc, dCc, A_log, Dco,
                                                h0, y);
}